// TripletLoss_24318104830758
// MI455X (gfx1250) — compile-verified
//
#include <hip/hip_runtime.h>
#include <math.h>

#define NN 512
#define DD 512
#define TRIPLET_MARGIN 0.2f

typedef float v2f __attribute__((ext_vector_type(2)));
typedef float v8f __attribute__((ext_vector_type(8)));

// ---------------------------------------------------------------------------
// Kernel 1: G = E * E^T  via V_WMMA_F32_16X16X4_F32 (full f32 precision).
// One wave per 16x16 output tile. 32x32 = 1024 tiles total.
//
// f32 A-matrix 16x4 layout (per ISA 7.12.2): lane l holds row M=l%16,
//   VGPR0 = K=(l<16?0:2), VGPR1 = K=(l<16?1:3).
// f32 B-matrix 4x16: lane l holds column N=l%16 with the same K pairing,
//   and B[k][n] = E[n][k], so A and B fragment loads are symmetric.
// C/D 16x16 f32: VGPR v, lane l -> M = v + (l<16?0:8), N = l%16.
// ---------------------------------------------------------------------------
__global__ __launch_bounds__(256) void gram_kernel(const float* __restrict__ E,
                                                   float* __restrict__ G) {
    const int lane  = threadIdx.x & 31;
    const int wave  = blockIdx.x * (blockDim.x >> 5) + (threadIdx.x >> 5);
    const int tm    = wave >> 5;          // tile row   (0..31)
    const int tn    = wave & 31;          // tile col   (0..31)
    const int r     = lane & 15;          // row within tile
    const int khalf = (lane >> 4) << 1;   // 0 or 2

    const float* __restrict__ arow = E + (size_t)(tm * 16 + r) * DD;
    const float* __restrict__ brow = E + (size_t)(tn * 16 + r) * DD;

    v8f acc = {};
#pragma unroll 8
    for (int k0 = 0; k0 < DD; k0 += 4) {
        v2f a = *(const v2f*)(arow + k0 + khalf);   // 8B aligned (k0+khalf even)
        v2f b = *(const v2f*)(brow + k0 + khalf);
        acc = __builtin_amdgcn_wmma_f32_16x16x4_f32(
            /*neg_a=*/false, a, /*neg_b=*/false, b,
            /*c_mod=*/(short)0, acc, /*reuse_a=*/false, /*reuse_b=*/false);
    }

    const int mbase = tm * 16 + ((lane >> 4) << 3);
    const int nn    = tn * 16 + r;
#pragma unroll
    for (int v = 0; v < 8; ++v) {
        G[(size_t)(mbase + v) * NN + nn] = acc[v];  // coalesced across lanes
    }
}

// ---------------------------------------------------------------------------
// Kernel 2: per-anchor semi-hard ("easy") negative mining + hinge accumulate.
// One 256-thread workgroup per anchor row a. pdist row lives in LDS.
// Faithfully reproduces the reference's masked-min/max shift formulas.
// ---------------------------------------------------------------------------
__global__ __launch_bounds__(256) void triplet_rows(const float* __restrict__ G,
                                                    const int* __restrict__ labels,
                                                    float* __restrict__ accum) {
    __shared__ float prow[NN];
    __shared__ int   lab[NN];
    __shared__ float redA[256];
    __shared__ float redB[256];

    const int a   = blockIdx.x;
    const int tid = threadIdx.x;

    const float gaa = G[(size_t)a * NN + a];
    for (int k = tid; k < NN; k += 256) {
        float d2 = gaa + G[(size_t)k * NN + k] - 2.0f * G[(size_t)a * NN + k];
        d2 = fmaxf(d2, 0.0f);
        // reference: zero-masked sqrt -> 0 where d2 == 0, sqrt(d2) otherwise
        prow[k] = (d2 == 0.0f) ? 0.0f : sqrtf(d2);
        lab[k]  = labels[k];
    }
    __syncthreads();

    const int la = lab[a];

    // --- block reduce: row max and row min over all k (reference axis_max/min)
    float lmax = -INFINITY, lmin = INFINITY;
    for (int k = tid; k < NN; k += 256) {
        lmax = fmaxf(lmax, prow[k]);
        lmin = fminf(lmin, prow[k]);
    }
    redA[tid] = lmax; redB[tid] = lmin;
    __syncthreads();
    for (int s = 128; s > 0; s >>= 1) {
        if (tid < s) {
            redA[tid] = fmaxf(redA[tid], redA[tid + s]);
            redB[tid] = fminf(redB[tid], redB[tid + s]);
        }
        __syncthreads();
    }
    const float rmax = redA[0];
    const float rmin = redB[0];
    __syncthreads();

    // --- neg_inside = axis_min + max_k (prow[k]-axis_min) * adjf[k]
    float lni = -INFINITY;
    for (int k = tid; k < NN; k += 256) {
        float v = (lab[k] != la) ? (prow[k] - rmin) : 0.0f;
        lni = fmaxf(lni, v);
    }
    redA[tid] = lni;
    __syncthreads();
    for (int s = 128; s > 0; s >>= 1) {
        if (tid < s) redA[tid] = fmaxf(redA[tid], redA[tid + s]);
        __syncthreads();
    }
    const float neg_inside = rmin + redA[0];
    __syncthreads();

    // --- per positive b: neg_outside = axis_max + min_k (prow[k]-axis_max)*mask
    float lsum = 0.0f;
    float pcnt = 0.0f;
    for (int b = tid; b < NN; b += 256) {
        if (lab[b] == la && b != a) {
            const float pab = prow[b];
            float minv = 0.0f;          // unmasked entries contribute 0 in ref
            int   any  = 0;
            for (int k = 0; k < NN; ++k) {   // LDS broadcast: all lanes same k
                if (lab[k] != la && prow[k] > pab) {
                    minv = fminf(minv, prow[k] - rmax);
                    any  = 1;
                }
            }
            const float shn = any ? (rmax + minv) : neg_inside;
            lsum += fmaxf(TRIPLET_MARGIN + pab - shn, 0.0f);
            pcnt += 1.0f;
        }
    }

    // --- block reduce loss sum + positive count, atomically accumulate
    redA[tid] = lsum; redB[tid] = pcnt;
    __syncthreads();
    for (int s = 128; s > 0; s >>= 1) {
        if (tid < s) {
            redA[tid] += redA[tid + s];
            redB[tid] += redB[tid + s];
        }
        __syncthreads();
    }
    if (tid == 0) {
        atomicAdd(&accum[0], redA[0]);
        atomicAdd(&accum[1], redB[0]);
    }
}

__global__ void init_accum(float* accum) {
    if (threadIdx.x < 2) accum[threadIdx.x] = 0.0f;
}

__global__ void finalize(const float* __restrict__ accum, float* __restrict__ out) {
    if (threadIdx.x == 0) out[0] = accum[0] / (accum[1] + 1e-8f);
}

// ---------------------------------------------------------------------------
extern "C" void kernel_launch(void* const* d_in, const int* in_sizes, int n_in,
                              void* d_out, int out_size, void* d_ws, size_t ws_size,
                              hipStream_t stream) {
    const int*   labels = (const int*)d_in[0];
    const float* emb    = (const float*)d_in[1];
    float*       out    = (float*)d_out;

    float* G     = (float*)d_ws;                                  // 1 MB
    float* accum = (float*)((char*)d_ws + (size_t)NN * NN * 4);   // 8 B

    init_accum<<<1, 32, 0, stream>>>(accum);
    // 1024 waves = 32x32 tiles; 8 waves (256 threads) per block -> 128 blocks
    gram_kernel<<<128, 256, 0, stream>>>(emb, G);
    triplet_rows<<<NN, 256, 0, stream>>>(G, labels, accum);
    finalize<<<1, 32, 0, stream>>>(accum, out);
}